// Model_20066087207170
// MI455X (gfx1250) — compile-verified
//
#include <hip/hip_runtime.h>
#include <math.h>

// Problem constants (reference): N=128, C=64, T=300, V=25, D=64
#define NROWS   38400      // N*T
#define VC      1600       // V*C == V*D
#define V25     25
#define T300    300
#define TR      16         // rows (M) per block = one WMMA M tile
#define BTHREADS 800       // 25 wave32 per block: one v-tile per wave (perfect balance)
#define XROW    1650       // LDS row stride in floats: 1600 + 1 pad DWORD per 32 stored
                           // (TDM pad_interval=4 -> 32 DWORDs, pad_amount=0 -> 1 DWORD)
                           // 1650 mod 64 = 50; 50*rl mod 64 distinct for rl=0..15 -> no conflicts
#define NTOT    61440000   // N*C*T*V
#define BN_EPS  1e-5f

typedef __attribute__((ext_vector_type(2))) float v2f;
typedef __attribute__((ext_vector_type(8))) float v8f;
typedef __attribute__((ext_vector_type(4))) unsigned int u32x4;
typedef __attribute__((ext_vector_type(4))) int i32x4;
typedef __attribute__((ext_vector_type(8))) int i32x8;

#if defined(__HIP_DEVICE_COMPILE__) && __has_builtin(__builtin_amdgcn_tensor_load_to_lds)
  #define HAVE_TDM 1
  #if defined(__clang_major__) && (__clang_major__ >= 23)
    #define TDM_6ARG 1        // amdgpu-toolchain (clang-23): 6-arg form
  #else
    #define TDM_6ARG 0        // ROCm 7.2 (clang-22): 5-arg form
  #endif
#else
  #define HAVE_TDM 0
#endif

// ---------------------------------------------------------------------------
// Kernel 0: zero the BN stat accumulators (ws is poisoned by the harness).
// ---------------------------------------------------------------------------
__global__ void k_zero(float* __restrict__ p, int n) {
    int i = blockIdx.x * blockDim.x + threadIdx.x;
    if (i < n) p[i] = 0.0f;
}

// ---------------------------------------------------------------------------
// Kernel 1: shift-gather + mask + GEMM (fp32 WMMA) + y store + BN partial sums.
// Block = 800 threads (25 wave32). Block owns 16 rows x all 25 v x 64 d;
// wave w owns v = w exactly (no tail imbalance).
// x0 rows are staged into LDS by the Tensor Data Mover (16 2D tiles:
// 64 lines x 25 contiguous floats, line stride 7500 floats), with TDM LDS
// padding of 1 DWORD per 32 stored -> row stride 1650 floats (bank-safe).
// Each wave: 4 fp32 WMMA accumulators, K=64 in 16 steps of
// V_WMMA_F32_16X16X4_F32 (4 N sub-tiles each -> 64 WMMAs).
// ---------------------------------------------------------------------------
__global__ __launch_bounds__(BTHREADS) void k_gemm_stats(
    const float* __restrict__ x0,      // [N,C,T,V]
    const float* __restrict__ W,       // [C,D] row-major
    const float* __restrict__ fmask,   // [V,C]
    const int*   __restrict__ shin,    // [1600]
    float* __restrict__ y,             // [NROWS, VC] workspace
    float* __restrict__ fsum,          // [1600]
    float* __restrict__ fsq)           // [1600]
{
    __shared__ float xs[TR * XROW];    // staged x rows, TDM source order, padded (105.6 KB)
    __shared__ float wlds[64 * 64];    // W tile (16 KB)
    __shared__ int   shoff[VC];        // shift_in -> padded LDS offset within a row
    __shared__ float fac[VC];          // tanh(feature_mask)+1 per destination (v,c)

    const int tid = threadIdx.x;
    const int m0  = blockIdx.x * TR;

#if HAVE_TDM
    // --- TDM: one 2D tensor_load_to_lds per row, issued by wave 0 (EXEC is
    // ignored by TDM; the branch is wave-uniform). Unrolled so the 16
    // tensor ops are statically visible and descriptor fields constant-fold.
    if (tid < 32) {
        // D# group1 (constant across rows):
        i32x8 g1 = {};
        g1[0] = (2 << 16)     // data_size = 4 bytes
              | (1 << 20)     // pad_enable
              | (4 << 22)     // pad_interval: 32 DWORDs between pads
              | (0 << 25);    // pad_amount: 1 DWORD
        g1[1] = (int)(25u << 16);   // tensor_dim0[15:0] at bits[63:48] -> 25
        g1[2] = (int)(64u << 16);   // tensor_dim0 hi = 0 | tensor_dim1 lo -> 64
        g1[3] = (int)(25u << 16);   // tensor_dim1 hi = 0 | tile_dim0 -> 25
        g1[4] = 64;                 // tile_dim1 = 64, tile_dim2 = 0 (2D)
        g1[5] = 7500;               // tensor_dim0_stride lo32 = 7500 elements
        g1[6] = 0;                  // stride hi | tensor_dim1_stride lo (unused, 2D)
        g1[7] = 0;
        i32x4 g2 = {};              // 2D: unused
        i32x4 g3 = {};
        typedef __attribute__((address_space(3))) float lds_float;
        unsigned int lds0 = (unsigned int)(uintptr_t)(lds_float*)&xs[0];
        #pragma unroll
        for (int rl = 0; rl < TR; ++rl) {
            int row = m0 + rl;
            int n = row / T300;
            int t = row - n * T300;
            unsigned long long ga =
                (unsigned long long)(uintptr_t)(x0 + (size_t)n * 480000 + (size_t)t * 25);
            u32x4 g0;
            g0[0] = 1u;                                        // count=1, user descriptor
            g0[1] = lds0 + (unsigned int)(rl * XROW * 4);      // lds_addr (bytes)
            g0[2] = (unsigned int)(ga & 0xFFFFFFFFu);          // global_addr[31:0]
            g0[3] = (unsigned int)((ga >> 32) & 0x01FFFFFFu)   // global_addr[56:32]
                  | (2u << 30);                                // type = 2 ("image")
#if TDM_6ARG
            i32x8 g4 = {};
            __builtin_amdgcn_tensor_load_to_lds(g0, g1, g2, g3, g4, 0);
#else
            __builtin_amdgcn_tensor_load_to_lds(g0, g1, g2, g3, 0);
#endif
        }
    }
#endif

    // --- stage W (coalesced) ---
    for (int i = tid; i < 64 * 64; i += BTHREADS) wlds[i] = W[i];

    // --- stage gather offsets + mask factors ---
    // shift_in[f] = src feature v_src*64 + c_src; TDM stored it at LDS linear
    // index l = c_src*25 + v_src, plus 1 pad DWORD per 32 stored -> l + l/32.
    for (int f = tid; f < VC; f += BTHREADS) {
        int s  = shin[f];
        int vs = s >> 6;
        int cs = s & 63;
        int l  = cs * V25 + vs;
        shoff[f] = l + (l >> 5);
        fac[f]   = tanhf(fmask[f]) + 1.0f;
    }

#if HAVE_TDM
    if (tid < 32) __builtin_amdgcn_s_wait_tensorcnt(0);
#else
    // --- fallback: scalar staging in TDM-identical LDS layout ---
    for (int idx = tid; idx < TR * VC; idx += BTHREADS) {
        int rl = idx / VC;
        int s  = idx - rl * VC;      // s = c*25 + v (x0 memory order, v fastest)
        int c  = s / V25;
        int v  = s - c * V25;
        int row = m0 + rl;
        int n = row / T300;
        int t = row - n * T300;
        xs[rl * XROW + s + (s >> 5)] =
            x0[(size_t)n * 480000 + (size_t)c * 7500 + t * 25 + v];
    }
#endif
    __syncthreads();

    const int lane = tid & 31;
    const int v    = tid >> 5;             // wave id == v (25 waves, 25 joints)
    const int rl   = lane & 15;            // A: both half-waves carry M=0..15
    const int q0   = (lane < 16) ? 0 : 2;  // A/B: K sub-offset per half-wave
    const int col  = lane & 15;            // B/C/D: N = col (per 16-lane half)

    v8f acc0 = {}, acc1 = {}, acc2 = {}, acc3 = {};

    #pragma unroll
    for (int k0 = 0; k0 < 64; k0 += 4) {
        // A 16x4 tile: gathered + masked from LDS
        int f0 = v * 64 + k0 + q0;
        v2f a;
        a.x = xs[rl * XROW + shoff[f0]]     * fac[f0];
        a.y = xs[rl * XROW + shoff[f0 + 1]] * fac[f0 + 1];

        // B 4x16 tiles for the 4 N sub-tiles
        int wr0 = (k0 + q0) * 64 + col;
        int wr1 = wr0 + 64;
        v2f b0, b1, b2, b3;
        b0.x = wlds[wr0 +  0]; b0.y = wlds[wr1 +  0];
        b1.x = wlds[wr0 + 16]; b1.y = wlds[wr1 + 16];
        b2.x = wlds[wr0 + 32]; b2.y = wlds[wr1 + 32];
        b3.x = wlds[wr0 + 48]; b3.y = wlds[wr1 + 48];

        acc0 = __builtin_amdgcn_wmma_f32_16x16x4_f32(false, a, false, b0, (short)0, acc0, false, false);
        acc1 = __builtin_amdgcn_wmma_f32_16x16x4_f32(false, a, false, b1, (short)0, acc1, false, false);
        acc2 = __builtin_amdgcn_wmma_f32_16x16x4_f32(false, a, false, b2, (short)0, acc2, false, false);
        acc3 = __builtin_amdgcn_wmma_f32_16x16x4_f32(false, a, false, b3, (short)0, acc3, false, false);
    }

    // Store y (coalesced 16-float segments) + per-feature partial sums.
    // C/D layout: vgpr i -> M = i + (lane<16 ? 0 : 8), N = col.
    const int mofs = (lane < 16) ? 0 : 8;
    v8f accs[4] = {acc0, acc1, acc2, acc3};
    #pragma unroll
    for (int dt = 0; dt < 4; ++dt) {
        int fbase = v * 64 + dt * 16 + col;
        float s1 = 0.0f, s2 = 0.0f;
        #pragma unroll
        for (int i = 0; i < 8; ++i) {
            float val = accs[dt][i];
            int row = m0 + i + mofs;
            y[(size_t)row * VC + fbase] = val;
            s1 += val;
            s2 += val * val;
        }
        // fold lanes 16-31 (M=8..15) into lanes 0-15 -> sum over all 16 rows
        s1 += __shfl_xor(s1, 16, 32);
        s2 += __shfl_xor(s2, 16, 32);
        if (lane < 16) {
            atomicAdd(&fsum[fbase], s1);
            atomicAdd(&fsq [fbase], s2);
        }
    }
}

// ---------------------------------------------------------------------------
// Kernel 2: fold BN stats (permuted through shift_out) + gamma/beta into
// per-feature scale/shift. (Linear bias b is a per-feature constant and is
// exactly cancelled by the mean subtraction, so it is omitted.)
// ---------------------------------------------------------------------------
__global__ void k_bnparams(
    const float* __restrict__ fsum, const float* __restrict__ fsq,
    const float* __restrict__ gamma, const float* __restrict__ beta,
    const int* __restrict__ shout,
    float* __restrict__ scale, float* __restrict__ shiftv)
{
    int f = blockIdx.x * blockDim.x + threadIdx.x;
    if (f >= VC) return;
    int sf = shout[f];
    const float inv = 1.0f / (float)NROWS;
    float mean = fsum[sf] * inv;
    float var  = fsq[sf] * inv - mean * mean;
    float sc   = gamma[f] * rsqrtf(var + BN_EPS);
    scale[f]  = sc;
    shiftv[f] = beta[f] - mean * sc;
}

// ---------------------------------------------------------------------------
// Kernel 3: BN apply + shift_out gather + residual + ReLU, output in [N,D,T,V].
// float4 on the two coalesced 246 MB streams (x0, out); the y gather stays
// L2-resident per n (1.9 MB working set << 192 MB L2).
// ---------------------------------------------------------------------------
__global__ __launch_bounds__(256) void k_apply(
    const float* __restrict__ y, const float* __restrict__ x0,
    const int* __restrict__ shout,
    const float* __restrict__ scale, const float* __restrict__ shiftv,
    float* __restrict__ out)
{
    size_t q = ((size_t)blockIdx.x * blockDim.x + threadIdx.x) * 4;
    if (q >= (size_t)NTOT) return;     // NTOT % 1024 == 0: no partial vectors
    float4 xin = *(const float4*)(x0 + q);
    float4 r;
    float* xi = &xin.x;
    float* rr = &r.x;
    #pragma unroll
    for (int i = 0; i < 4; ++i) {
        size_t o = q + (size_t)i;
        int v = (int)(o % 25);
        size_t t1 = o / 25;
        int t = (int)(t1 % T300);
        size_t t2 = t1 / T300;
        int d = (int)(t2 % 64);
        int n = (int)(t2 / 64);
        int row = n * T300 + t;
        int f = v * 64 + d;
        float val = y[(size_t)row * VC + shout[f]] * scale[f] + shiftv[f] + xi[i];
        rr[i] = fmaxf(val, 0.0f);
    }
    *(float4*)(out + q) = r;
}

// ---------------------------------------------------------------------------
// Host launch: inputs in setup_inputs() order:
// 0:x0  1:W  2:b(unused, cancelled by BN)  3:feature_mask  4:gamma  5:beta
// 6:shift_in  7:shift_out
// ws layout (floats): y[61440000] | fsum[1600] | fsq[1600] | scale[1600] | shift[1600]
// ---------------------------------------------------------------------------
extern "C" void kernel_launch(void* const* d_in, const int* in_sizes, int n_in,
                              void* d_out, int out_size, void* d_ws, size_t ws_size,
                              hipStream_t stream)
{
    const float* x0    = (const float*)d_in[0];
    const float* W     = (const float*)d_in[1];
    const float* fmask = (const float*)d_in[3];
    const float* gamma = (const float*)d_in[4];
    const float* beta  = (const float*)d_in[5];
    const int*   shin  = (const int*)d_in[6];
    const int*   shout = (const int*)d_in[7];

    float* ws     = (float*)d_ws;
    float* y      = ws;                      // 61,440,000 floats
    float* fsum   = ws + (size_t)NTOT;       // 1600
    float* fsq    = fsum + VC;               // 1600
    float* scale  = fsq + VC;                // 1600
    float* shiftv = scale + VC;              // 1600
    float* out    = (float*)d_out;

    k_zero<<<(2 * VC + 255) / 256, 256, 0, stream>>>(fsum, 2 * VC);
    k_gemm_stats<<<NROWS / TR, BTHREADS, 0, stream>>>(x0, W, fmask, shin, y, fsum, fsq);
    k_bnparams<<<(VC + 255) / 256, 256, 0, stream>>>(fsum, fsq, gamma, beta, shout, scale, shiftv);
    k_apply<<<(NTOT / 4 + 255) / 256, 256, 0, stream>>>(y, x0, shout, scale, shiftv, out);
}